// Turb2dPDE_4887672782852
// MI455X (gfx1250) — compile-verified
//
#include <hip/hip_runtime.h>
#include <math.h>

// ---------------------------------------------------------------------------
// 2D turbulence PDE RHS for MI455X (gfx1250, wave32, WMMA).
// Poisson solve done as dense 65-point DFTs via V_WMMA_F32_16X16X4_F32.
// ---------------------------------------------------------------------------

typedef __attribute__((ext_vector_type(2))) float v2f;
typedef __attribute__((ext_vector_type(8))) float v8f;

#define N64   64
#define NF    65          // padded (wrap) DFT size
#define M80   80          // 5 x 16 tile-padded matrix dim
#define K68   68          // K padded to multiple of 4 (cols 65..67 are zero)
#define LDB   83          // LDS row stride (odd -> conflict-free strided loads)
#define LDX   65          // LDS stride for 64x64 tiles
#define TWO_PI 6.283185307179586f
#define DX0   (3.14159265358979323846f / 64.0f)

// ---------------------------------------------------------------------------
// Setup: DFT tables Cm[i][n]=cos(2*pi*i*n/65), Sm[i][n]=sin(...), zero-padded
// to 80x80.  Symmetric matrices => transposes are free.
// ---------------------------------------------------------------------------
__global__ void dft_tables_kernel(float* __restrict__ gCm, float* __restrict__ gSm) {
    int idx = blockIdx.x * blockDim.x + threadIdx.x;
    if (idx >= M80 * M80) return;
    int r = idx / M80, c = idx % M80;
    float cv = 0.0f, sv = 0.0f;
    if (r < NF && c < NF) {
        int m = (r * c) % NF;                 // exact phase reduction
        float th = TWO_PI * (float)m / (float)NF;
        cv = cosf(th);
        sv = sinf(th);
    }
    gCm[idx] = cv;
    gSm[idx] = sv;
}

// ---------------------------------------------------------------------------
// One 16x16 output tile accumulated over K=68 with V_WMMA_F32_16X16X4_F32.
// A: MxK (row-major, stride lda), sign asign folded into fragment loads.
// B: KxN (row-major, stride ldb), sign bsign folded into fragment loads.
// f32 A-frag layout (ISA 7.12.2): lanes 0-15 = M rows, vgpr pair = {K, K+1},
// lanes 16-31 hold K+2/K+3.  B mirrors with N striped across lanes.
// ---------------------------------------------------------------------------
__device__ __forceinline__ v8f mm_tile(const float* A, int lda, float asign,
                                       const float* B, int ldb, float bsign,
                                       int tm, int tn, v8f acc) {
    const int lane = threadIdx.x & 31;
    const int mrow = tm * 16 + (lane & 15);
    const int ncol = tn * 16 + (lane & 15);
    const int kh   = (lane >> 4) * 2;
    for (int k = 0; k < K68; k += 4) {
        const int kk = k + kh;
        v2f a, b;
        a.x = asign * A[mrow * lda + kk];
        a.y = asign * A[mrow * lda + kk + 1];
        b.x = bsign * B[kk * ldb + ncol];
        b.y = bsign * B[(kk + 1) * ldb + ncol];
        acc = __builtin_amdgcn_wmma_f32_16x16x4_f32(
            /*neg_a=*/false, a, /*neg_b=*/false, b,
            /*c_mod=*/(short)0, acc, /*reuse_a=*/false, /*reuse_b=*/false);
    }
    return acc;
}

// C/D layout (ISA 7.12.2): vgpr r, lanes 0-15 -> M = r, lanes 16-31 -> M = r+8.
__device__ __forceinline__ void store_tile(float* D, int ldd, int tm, int tn,
                                           v8f acc, float scale) {
    const int lane  = threadIdx.x & 31;
    const int n     = tn * 16 + (lane & 15);
    const int mbase = tm * 16 + (lane >> 4) * 8;
#pragma unroll
    for (int r = 0; r < 8; ++r)
        D[(mbase + r) * ldd + n] = acc[r] * scale;
}

// ---------------------------------------------------------------------------
// Main kernel: one workgroup (8 wave32) per batch element.
// ---------------------------------------------------------------------------
__global__ void __launch_bounds__(256, 1) turb2d_rhs_kernel(
    const float* __restrict__ y0, const int* __restrict__ env,
    const float* __restrict__ params, const float* __restrict__ domain,
    const float* __restrict__ gCm, const float* __restrict__ gSm,
    float* __restrict__ out)
{
    extern __shared__ float smem[];
    float* bufA = smem;                 // 80x83 : w  -> Wim -> Uim
    float* bufB = bufA + M80 * LDB;     // 80x83 : Tre -> Sre
    float* bufC = bufB + M80 * LDB;     // 80x83 : Tim -> Sim
    float* bufD = bufC + M80 * LDB;     // 80x83 : Wre -> Ure
    float* sX   = bufD + M80 * LDB;     // 64x65 : vorticity
    float* sPsi = sX + N64 * LDX;       // 64x65 : streamfunction

    const int b   = blockIdx.x;
    const int tid = threadIdx.x;
    const int wid = tid >> 5;

    // per-batch scalars (uniform -> scalar loads)
    const int   e        = env[b];
    const float fdx      = domain[e] * DX0;
    const float fdx2     = fdx * fdx;
    const float inv_fdx2 = 1.0f / fdx2;
    const float mu       = params[2 * e];

    // ---- phase 0: load x into LDS -----------------------------------------
    const float* xg = y0 + (size_t)b * (N64 * N64);
    for (int i = tid; i < N64 * N64; i += 256)
        sX[(i >> 6) * LDX + (i & 63)] = xg[i];
    __syncthreads();

    // ---- phase 1: w80 = -x, wrap-padded 65x65, zero-padded to 80x80 -------
    for (int i = tid; i < M80 * M80; i += 256) {
        int r = i / M80, c = i % M80;
        float v = 0.0f;
        if (r < NF && c < NF) {
            int rr = (r == 64) ? 0 : r;
            int cc = (c == 64) ? 0 : c;
            v = -sX[rr * LDX + cc];
        }
        bufA[r * LDB + c] = v;
    }
    __syncthreads();

    // ---- phase 2: T = F * w, F = C - i*S  =>  Tre = C*w, Tim = -S*w -------
    for (int t = wid; t < 25; t += 8) {
        int tm = t / 5, tn = t % 5;
        v8f acc = {};
        acc = mm_tile(gCm, M80, 1.0f, bufA, LDB, 1.0f, tm, tn, acc);
        store_tile(bufB, LDB, tm, tn, acc, 1.0f);
        v8f acc2 = {};
        acc2 = mm_tile(gSm, M80, -1.0f, bufA, LDB, 1.0f, tm, tn, acc2);
        store_tile(bufC, LDB, tm, tn, acc2, 1.0f);
    }
    __syncthreads();

    // ---- phase 3: W = T * F^T  (C,S symmetric => F^T loads are plain) -----
    // Wre = Tre*C + Tim*S ; Wim = Tim*C - Tre*S
    for (int t = wid; t < 25; t += 8) {
        int tm = t / 5, tn = t % 5;
        v8f acc = {};
        acc = mm_tile(bufB, LDB, 1.0f, gCm, M80, 1.0f, tm, tn, acc);
        acc = mm_tile(bufC, LDB, 1.0f, gSm, M80, 1.0f, tm, tn, acc);
        store_tile(bufD, LDB, tm, tn, acc, 1.0f);
        v8f acc2 = {};
        acc2 = mm_tile(bufC, LDB, 1.0f, gCm, M80, 1.0f, tm, tn, acc2);
        acc2 = mm_tile(bufB, LDB, 1.0f, gSm, M80, -1.0f, tm, tn, acc2);
        store_tile(bufA, LDB, tm, tn, acc2, 1.0f);
    }
    __syncthreads();

    // ---- phase 4: spectral filter U = G .* W ------------------------------
    // G = -1/(1e-12 + (kx^2+ky^2)/fdx^2), G[0][0] = 0, k = 2*pi*fftfreq(65)
    for (int i = tid; i < M80 * M80; i += 256) {
        int r = i / M80, c = i % M80;
        float g = 0.0f;
        if (r < NF && c < NF && !(r == 0 && c == 0)) {
            float fr = (float)((r <= 32) ? r : r - 65) * (1.0f / 65.0f);
            float fc = (float)((c <= 32) ? c : c - 65) * (1.0f / 65.0f);
            float kx = TWO_PI * fr, ky = TWO_PI * fc;
            g = -1.0f / (1e-12f + (kx * kx + ky * ky) * inv_fdx2);
        }
        bufD[r * LDB + c] *= g;   // Ure
        bufA[r * LDB + c] *= g;   // Uim
    }
    __syncthreads();

    // ---- phase 5: Sp = F* * U, F* = C + i*S -------------------------------
    // Sre = C*Ure - S*Uim ; Sim = C*Uim + S*Ure
    for (int t = wid; t < 25; t += 8) {
        int tm = t / 5, tn = t % 5;
        v8f acc = {};
        acc = mm_tile(gCm, M80, 1.0f, bufD, LDB, 1.0f, tm, tn, acc);
        acc = mm_tile(gSm, M80, -1.0f, bufA, LDB, 1.0f, tm, tn, acc);
        store_tile(bufB, LDB, tm, tn, acc, 1.0f);
        v8f acc2 = {};
        acc2 = mm_tile(gCm, M80, 1.0f, bufA, LDB, 1.0f, tm, tn, acc2);
        acc2 = mm_tile(gSm, M80, 1.0f, bufD, LDB, 1.0f, tm, tn, acc2);
        store_tile(bufC, LDB, tm, tn, acc2, 1.0f);
    }
    __syncthreads();

    // ---- phase 6: psi = Re{Sp * F*^T}/65^2, crop to 64x64 -----------------
    // psi = (Sre*C - Sim*S)/4225  (only the 4x4 tile block covering 0..63)
    for (int t = wid; t < 16; t += 8) {
        int tm = t / 4, tn = t % 4;
        v8f acc = {};
        acc = mm_tile(bufB, LDB, 1.0f, gCm, M80, 1.0f, tm, tn, acc);
        acc = mm_tile(bufC, LDB, -1.0f, gSm, M80, 1.0f, tm, tn, acc);
        store_tile(sPsi, LDX, tm, tn, acc, 1.0f / 4225.0f);
    }
    __syncthreads();

    // ---- phase 7: Arakawa Jacobian + diffusion stencil --------------------
    const float inv4 = 0.25f * inv_fdx2;
    float* og = out + (size_t)b * (N64 * N64);
    for (int idx = tid; idx < N64 * N64; idx += 256) {
        int i = idx >> 6, j = idx & 63;
        int ip = (i + 1) & 63, im = (i - 1) & 63;
        int jp = (j + 1) & 63, jm = (j - 1) & 63;

        float xc  = sX[i  * LDX + j];
        float xr  = sX[i  * LDX + jp], xl  = sX[i  * LDX + jm];
        float xd  = sX[ip * LDX + j],  xu  = sX[im * LDX + j];
        float xdr = sX[ip * LDX + jp], xdl = sX[ip * LDX + jm];
        float xur = sX[im * LDX + jp], xul = sX[im * LDX + jm];

        float pr  = sPsi[i  * LDX + jp], pl  = sPsi[i  * LDX + jm];
        float pd  = sPsi[ip * LDX + j],  pu  = sPsi[im * LDX + j];
        float pdr = sPsi[ip * LDX + jp], pdl = sPsi[ip * LDX + jm];
        float pur = sPsi[im * LDX + jp], pul = sPsi[im * LDX + jm];

        float J1 = ((pd - pu) * (xr - xl) - (pr - pl) * (xd - xu)) * inv4;
        float J2 = (xr * (pdr - pur) - xl * (pdl - pul)
                  - xd * (pdr - pdl) + xu * (pur - pul)) * inv4;
        float J3 = (xdr * (pd - pr) - xul * (pl - pu)
                  - xdl * (pd - pl) + xur * (pr - pu)) * inv4;

        float lap = xu + xl - 4.0f * xc + xr + xdl + xd;

        og[idx] = -(J1 + J2 + J3) * (1.0f / 3.0f) + mu * lap * inv_fdx2;
    }
}

// ---------------------------------------------------------------------------
// Launch.  d_in order: t, y0, env, codes, params, domain.
// ---------------------------------------------------------------------------
extern "C" void kernel_launch(void* const* d_in, const int* in_sizes, int n_in,
                              void* d_out, int out_size, void* d_ws, size_t ws_size,
                              hipStream_t stream) {
    const float* y0     = (const float*)d_in[1];
    const int*   env    = (const int*)d_in[2];
    const float* params = (const float*)d_in[4];
    const float* domain = (const float*)d_in[5];
    float*       out    = (float*)d_out;

    float* gCm = (float*)d_ws;           // 80*80 floats
    float* gSm = gCm + M80 * M80;        // 80*80 floats

    const int B = in_sizes[1] / (N64 * N64);

    dft_tables_kernel<<<(M80 * M80 + 255) / 256, 256, 0, stream>>>(gCm, gSm);

    const size_t smem_bytes =
        (size_t)(4 * M80 * LDB + 2 * N64 * LDX) * sizeof(float);  // ~136 KB
    (void)hipFuncSetAttribute((const void*)turb2d_rhs_kernel,
                              hipFuncAttributeMaxDynamicSharedMemorySize,
                              (int)smem_bytes);

    turb2d_rhs_kernel<<<B, 256, smem_bytes, stream>>>(
        y0, env, params, domain, gCm, gSm, out);
}